// GCPMessagePassing_32083405701170
// MI455X (gfx1250) — compile-verified
//
#include <hip/hip_runtime.h>
#include <math.h>

// ---------------------------------------------------------------------------
// GCP message passing, fused per-edge-tile kernel for gfx1250 (MI455X).
// One wave32 handles 16 edges. ALL GEMMs (the two big s_out GEMMs, the two
// gate GEMMs, and the four vector-path GEMMs with M=48=(16 edges x 3 coords))
// run on V_WMMA_F32_16X16X4_F32 with f32 accumulation.
// B-operand weights are pre-packed once per launch into d_ws as [K/4][N][4]
// (zero-padded K) so each lane's B pair is a single global_load_b64; a
// template fallback keeps clamped b32 loads if ws_size is too small.
// ---------------------------------------------------------------------------

typedef float v2f __attribute__((ext_vector_type(2)));
typedef float v4f __attribute__((ext_vector_type(4)));
typedef float v8f __attribute__((ext_vector_type(8)));

#define N_NODES  10000
#define N_EDGES  160000
#define OUT_CH   176          // 128 scalar + 16*3 vector
#define OUT_ELTS (N_NODES * OUT_CH)

// packed-weight layout inside d_ws (float indices)
#define P1_OFF   0            // g1_Wso packed [84][128][4]   (K 333 -> 336)
#define P2_OFF   43008        // g2_Wso packed [39][128][4]   (K 153 -> 156)
#define PG1_OFF  62976        // g1_Wg  packed [32][16][4]    (K 128)
#define PG2_OFF  65024        // g2_Wg  packed [32][16][4]
#define PW_FLOATS 67072
#define PW_BYTES  (PW_FLOATS * 4)   // 268288 bytes

__device__ __forceinline__ float sigm(float x) { return 1.0f / (1.0f + expf(-x)); }

__global__ void zero_kernel(float* __restrict__ p, int n) {
    int i = blockIdx.x * blockDim.x + threadIdx.x;
    if (i < n) p[i] = 0.0f;
}

__global__ void pack_weights(const float* __restrict__ Wso1,  // [333,128]
                             const float* __restrict__ Wso2,  // [153,128]
                             const float* __restrict__ Wg1,   // [128,16]
                             const float* __restrict__ Wg2,   // [128,16]
                             float* __restrict__ pw) {
    int i = blockIdx.x * blockDim.x + threadIdx.x;
    if (i < P2_OFF) {                       // g1_Wso
        int kb = i >> 9, rem = i & 511, n = rem >> 2, r = rem & 3, k = kb * 4 + r;
        pw[i] = (k < 333) ? Wso1[k * 128 + n] : 0.0f;
    } else if (i < PG1_OFF) {               // g2_Wso
        int j = i - P2_OFF;
        int kb = j >> 9, rem = j & 511, n = rem >> 2, r = rem & 3, k = kb * 4 + r;
        pw[i] = (k < 153) ? Wso2[k * 128 + n] : 0.0f;
    } else if (i < PG2_OFF) {               // g1_Wg
        int j = i - PG1_OFF;
        int kb = j >> 6, rem = j & 63, n = rem >> 2, r = rem & 3, k = kb * 4 + r;
        pw[i] = Wg1[k * 16 + n];
    } else if (i < PW_FLOATS) {             // g2_Wg
        int j = i - PG2_OFF;
        int kb = j >> 6, rem = j & 63, n = rem >> 2, r = rem & 3, k = kb * 4 + r;
        pw[i] = Wg2[k * 16 + n];
    }
}

template <bool PACKED>
__global__ __launch_bounds__(32)
void gcp_edge_kernel(const float* __restrict__ node_s,   // [N,128]
                     const float* __restrict__ node_v,   // [N,16,3]
                     const float* __restrict__ edge_s,   // [E,32]
                     const float* __restrict__ edge_v,   // [E,4,3]
                     const float* __restrict__ frames,   // [E,3,3]
                     const float* __restrict__ g1_Wdown, // [36,36]
                     const float* __restrict__ g1_Wdf,   // [36,3]
                     const float* __restrict__ g1_Wso,   // [333,128]
                     const float* __restrict__ g1_bso,   // [128]
                     const float* __restrict__ g1_Wup,   // [36,16]
                     const float* __restrict__ g1_Wg,    // [128,16]
                     const float* __restrict__ g1_bg,    // [16]
                     const float* __restrict__ g2_Wdown, // [16,16]
                     const float* __restrict__ g2_Wdf,   // [16,3]
                     const float* __restrict__ g2_Wso,   // [153,128]
                     const float* __restrict__ g2_bso,   // [128]
                     const float* __restrict__ g2_Wup,   // [16,16]
                     const float* __restrict__ g2_Wg,    // [128,16]
                     const float* __restrict__ g2_bg,    // [16]
                     const float* __restrict__ att_W,    // [128,1]
                     const float* __restrict__ att_b,    // [1]
                     const int*   __restrict__ edge_index, // [2,E]
                     const float* __restrict__ pw,       // packed weights (d_ws)
                     float* __restrict__ out)            // [N,176]
{
    // ---- LDS staging (one wave per block; total ~61 KB) ----
    __shared__ __align__(16) float sm_merged1[16][336];   // [s 0:288 | vnorm 288:324 | local 324:333 | pad]
    __shared__ __align__(16) float sm_vpre[16][3][36];    // v_pre rows (e,x), K=36
    __shared__ __align__(16) float sm_vhid1[16][3][36];   // rows (e,x), cols j
    __shared__ __align__(16) float sm_merged2[16][160];   // [silu_s1 0:128 | vnorm2 128:144 | local2 144:153 | pad]
    __shared__ __align__(16) float sm_vpre2[16][3][16];   // gated v_out1, rows (e,x), cols vo
    __shared__ __align__(16) float sm_vhid2[16][3][16];
    __shared__ __align__(16) float sm_s2[16][128];        // silu(s_out2)
    __shared__ __align__(16) float sm_gate[16][16];
    __shared__ float sm_attn[16];
    __shared__ int   sm_row[16];
    __shared__ int   sm_col[16];

    const int lane = threadIdx.x;        // 0..31
    const int e0   = blockIdx.x * 16;    // first edge of tile
    const int half = lane >> 4;          // 0 or 1 (WMMA lane-half)
    const int lc   = lane & 15;          // column / A-row within tile
    const v8f vz   = {0.f, 0.f, 0.f, 0.f, 0.f, 0.f, 0.f, 0.f};

    const float* vpre_f  = &sm_vpre[0][0][0];   // [48][36]
    const float* vhid1_f = &sm_vhid1[0][0][0];  // [48][36]
    const float* vpre2_f = &sm_vpre2[0][0][0];  // [48][16]
    const float* vhid2_f = &sm_vhid2[0][0][0];  // [48][16]

    // ---- indices ----
    if (lane < 16) {
        sm_row[lane] = edge_index[e0 + lane];
        sm_col[lane] = edge_index[N_EDGES + e0 + lane];
    }
    __syncthreads();

    // ---- gather scalar message (b128): ms = [node_s[row] | edge_s | node_s[col]] ----
    for (int t = lane; t < 16 * 72; t += 32) {
        int e = t / 72, q = (t - e * 72) * 4;
        v4f v;
        if (q < 128)      v = *(const v4f*)&node_s[(size_t)sm_row[e] * 128 + q];
        else if (q < 160) v = *(const v4f*)&edge_s[(size_t)(e0 + e) * 32 + (q - 128)];
        else              v = *(const v4f*)&node_s[(size_t)sm_col[e] * 128 + (q - 160)];
        *(v4f*)&sm_merged1[e][q] = v;
    }
    // ---- gather vector message (b128 loads, transposed scatter to v_pre[(e,x)][k]) ----
    for (int t = lane; t < 16 * 27; t += 32) {
        int e = t / 27, q = (t - e * 27) * 4;
        v4f v;
        if (q < 48)      v = *(const v4f*)&node_v[(size_t)sm_row[e] * 48 + q];
        else if (q < 60) v = *(const v4f*)&edge_v[(size_t)(e0 + e) * 12 + (q - 48)];
        else             v = *(const v4f*)&node_v[(size_t)sm_col[e] * 48 + (q - 60)];
#pragma unroll
        for (int u = 0; u < 4; ++u) {
            int f = q + u, k = f / 3, x = f - k * 3;
            sm_vpre[e][x][k] = v[u];
        }
    }
    if (lane < 16) {                       // zero K-pad cols 333..335
        sm_merged1[lane][333] = 0.0f;
        sm_merged1[lane][334] = 0.0f;
        sm_merged1[lane][335] = 0.0f;
    }
    __syncthreads();

    // ---- GCP1: v_hid1[48,36] = v_pre[48,36] @ Wdown1[36,36] via WMMA (3x3 tiles) ----
    {
        v8f c9[3][3];
#pragma unroll
        for (int mt = 0; mt < 3; ++mt)
#pragma unroll
            for (int nt = 0; nt < 3; ++nt) c9[mt][nt] = vz;
        for (int k0 = 0; k0 < 36; k0 += 4) {
            const int kk = k0 + half * 2;
            v2f a[3], b[3];
#pragma unroll
            for (int mt = 0; mt < 3; ++mt)
                a[mt] = *(const v2f*)&vpre_f[(mt * 16 + lc) * 36 + kk];
#pragma unroll
            for (int nt = 0; nt < 3; ++nt) {
                const int n  = nt * 16 + lc;
                const int nc = (n < 36) ? n : 35;        // clamp; cols >=36 discarded
                b[nt].x = g1_Wdown[kk * 36 + nc];
                b[nt].y = g1_Wdown[(kk + 1) * 36 + nc];
            }
#pragma unroll
            for (int mt = 0; mt < 3; ++mt)
#pragma unroll
                for (int nt = 0; nt < 3; ++nt)
                    c9[mt][nt] = __builtin_amdgcn_wmma_f32_16x16x4_f32(
                        false, a[mt], false, b[nt], (short)0, c9[mt][nt], false, false);
        }
#pragma unroll
        for (int mt = 0; mt < 3; ++mt)
#pragma unroll
            for (int nt = 0; nt < 3; ++nt) {
                const int n = nt * 16 + lc;
                if (n < 36) {
#pragma unroll
                    for (int r = 0; r < 8; ++r) {
                        int grow = mt * 16 + r + half * 8;
                        int e = grow / 3, x = grow - e * 3;
                        sm_vhid1[e][x][n] = c9[mt][nt][r];
                    }
                }
            }
    }
    __syncthreads();

    // ---- v_norm1 -> merged1[288+j] ----
    for (int t = lane; t < 16 * 36; t += 32) {
        int e = t / 36, j = t - e * 36;
        float a = sm_vhid1[e][0][j], b = sm_vhid1[e][1][j], c = sm_vhid1[e][2][j];
        sm_merged1[e][288 + j] = sqrtf(a * a + b * b + c * c + 1e-8f);
    }
    // ---- local1 = (frames @ (v_pre @ Wdf1))^T -> merged1[324 + c*3 + a] ----
    for (int t = lane; t < 16 * 9; t += 32) {
        int e = t / 9, r = t - e * 9, c = r / 3, a = r - c * 3;
        float vf0 = 0.0f, vf1 = 0.0f, vf2 = 0.0f;
        for (int k = 0; k < 36; ++k) {
            float w = g1_Wdf[k * 3 + c];
            vf0 += sm_vpre[e][0][k] * w;
            vf1 += sm_vpre[e][1][k] * w;
            vf2 += sm_vpre[e][2][k] * w;
        }
        const float* fr = frames + (size_t)(e0 + e) * 9 + a * 3;
        sm_merged1[e][324 + c * 3 + a] = fr[0] * vf0 + fr[1] * vf1 + fr[2] * vf2;
    }
    __syncthreads();

    // ---- s_out1 = merged1 @ Wso1 + bso1 via WMMA (8 independent chains) ----
    {
        v8f c8[8];
#pragma unroll
        for (int nt = 0; nt < 8; ++nt) c8[nt] = vz;
        for (int k0 = 0; k0 < 336; k0 += 4) {
            const int kk = k0 + half * 2;
            v2f a = *(const v2f*)&sm_merged1[lc][kk];
#pragma unroll
            for (int nt = 0; nt < 8; ++nt) {
                const int n = nt * 16 + lc;
                v2f b;
                if (PACKED) {
                    b = *(const v2f*)&pw[P1_OFF + (k0 >> 2) * 512 + n * 4 + half * 2];
                } else {
                    const int kr0 = ((kk     < 333) ? kk     : 332) * 128;
                    const int kr1 = ((kk + 1 < 333) ? kk + 1 : 332) * 128;
                    b.x = g1_Wso[kr0 + n];
                    b.y = g1_Wso[kr1 + n];
                }
                c8[nt] = __builtin_amdgcn_wmma_f32_16x16x4_f32(false, a, false, b,
                                                               (short)0, c8[nt], false, false);
            }
        }
#pragma unroll
        for (int nt = 0; nt < 8; ++nt) {
            const int n = nt * 16 + lc;
            const float bias = g1_bso[n];
#pragma unroll
            for (int r = 0; r < 8; ++r) {
                const int m = r + half * 8;
                float x = c8[nt][r] + bias;
                sm_merged2[m][n] = x * sigm(x);                 // silu(s_out1)
            }
        }
    }
    __syncthreads();

    // ---- gate1 = sigmoid(silu(s_out1) @ Wg1 + bg1) via WMMA (4 split chains) ----
    {
        v8f part[4];
#pragma unroll
        for (int i = 0; i < 4; ++i) part[i] = vz;
        for (int kb = 0; kb < 128; kb += 16) {
#pragma unroll
            for (int i = 0; i < 4; ++i) {
                const int kk = kb + i * 4 + half * 2;
                v2f a = *(const v2f*)&sm_merged2[lc][kk];
                v2f b;
                if (PACKED) {
                    b = *(const v2f*)&pw[PG1_OFF + ((kb >> 2) + i) * 64 + lc * 4 + half * 2];
                } else {
                    b.x = g1_Wg[kk * 16 + lc];
                    b.y = g1_Wg[(kk + 1) * 16 + lc];
                }
                part[i] = __builtin_amdgcn_wmma_f32_16x16x4_f32(false, a, false, b,
                                                                (short)0, part[i], false, false);
            }
        }
        v8f g8 = (part[0] + part[1]) + (part[2] + part[3]);
        const float bias = g1_bg[lc];
#pragma unroll
        for (int r = 0; r < 8; ++r) {
            const int m = r + half * 8;
            sm_gate[m][lc] = sigm(g8[r] + bias);
        }
    }
    __syncthreads();

    // ---- v_out1[48,16] = v_hid1 @ Wup1, gated -> sm_vpre2 (GCP2 A operand) ----
    {
        v8f c3[3];
#pragma unroll
        for (int mt = 0; mt < 3; ++mt) c3[mt] = vz;
        for (int k0 = 0; k0 < 36; k0 += 4) {
            const int kk = k0 + half * 2;
            v2f b;
            b.x = g1_Wup[kk * 16 + lc];
            b.y = g1_Wup[(kk + 1) * 16 + lc];
#pragma unroll
            for (int mt = 0; mt < 3; ++mt) {
                v2f a = *(const v2f*)&vhid1_f[(mt * 16 + lc) * 36 + kk];
                c3[mt] = __builtin_amdgcn_wmma_f32_16x16x4_f32(false, a, false, b,
                                                               (short)0, c3[mt], false, false);
            }
        }
#pragma unroll
        for (int mt = 0; mt < 3; ++mt)
#pragma unroll
            for (int r = 0; r < 8; ++r) {
                int grow = mt * 16 + r + half * 8;
                int e = grow / 3, x = grow - e * 3;
                sm_vpre2[e][x][lc] = c3[mt][r] * sm_gate[e][lc];
            }
    }
    __syncthreads();

    // ---- GCP2: v_hid2[48,16] = v_pre2 @ Wdown2[16,16] via WMMA ----
    {
        v8f c3[3];
#pragma unroll
        for (int mt = 0; mt < 3; ++mt) c3[mt] = vz;
        for (int k0 = 0; k0 < 16; k0 += 4) {
            const int kk = k0 + half * 2;
            v2f b;
            b.x = g2_Wdown[kk * 16 + lc];
            b.y = g2_Wdown[(kk + 1) * 16 + lc];
#pragma unroll
            for (int mt = 0; mt < 3; ++mt) {
                v2f a = *(const v2f*)&vpre2_f[(mt * 16 + lc) * 16 + kk];
                c3[mt] = __builtin_amdgcn_wmma_f32_16x16x4_f32(false, a, false, b,
                                                               (short)0, c3[mt], false, false);
            }
        }
#pragma unroll
        for (int mt = 0; mt < 3; ++mt)
#pragma unroll
            for (int r = 0; r < 8; ++r) {
                int grow = mt * 16 + r + half * 8;
                int e = grow / 3, x = grow - e * 3;
                sm_vhid2[e][x][lc] = c3[mt][r];
            }
    }
    __syncthreads();

    // ---- v_norm2 / local2 -> merged2 ----
    for (int t = lane; t < 16 * 16; t += 32) {
        int e = t / 16, j = t - e * 16;
        float a = sm_vhid2[e][0][j], b = sm_vhid2[e][1][j], c = sm_vhid2[e][2][j];
        sm_merged2[e][128 + j] = sqrtf(a * a + b * b + c * c + 1e-8f);
    }
    for (int t = lane; t < 16 * 9; t += 32) {
        int e = t / 9, r = t - e * 9, c = r / 3, a = r - c * 3;
        float vf0 = 0.0f, vf1 = 0.0f, vf2 = 0.0f;
        for (int k = 0; k < 16; ++k) {
            float w = g2_Wdf[k * 3 + c];
            vf0 += sm_vpre2[e][0][k] * w;
            vf1 += sm_vpre2[e][1][k] * w;
            vf2 += sm_vpre2[e][2][k] * w;
        }
        const float* fr = frames + (size_t)(e0 + e) * 9 + a * 3;
        sm_merged2[e][144 + c * 3 + a] = fr[0] * vf0 + fr[1] * vf1 + fr[2] * vf2;
    }
    for (int t = lane; t < 16 * 7; t += 32) {  // zero K-pad cols 153..159
        int e = t / 7, c = 153 + (t - e * 7);
        sm_merged2[e][c] = 0.0f;
    }
    __syncthreads();

    // ---- s_out2 = merged2 @ Wso2 + bso2 via WMMA ----
    {
        v8f c8[8];
#pragma unroll
        for (int nt = 0; nt < 8; ++nt) c8[nt] = vz;
        for (int k0 = 0; k0 < 156; k0 += 4) {
            const int kk = k0 + half * 2;
            v2f a = *(const v2f*)&sm_merged2[lc][kk];
#pragma unroll
            for (int nt = 0; nt < 8; ++nt) {
                const int n = nt * 16 + lc;
                v2f b;
                if (PACKED) {
                    b = *(const v2f*)&pw[P2_OFF + (k0 >> 2) * 512 + n * 4 + half * 2];
                } else {
                    const int kr0 = ((kk     < 153) ? kk     : 152) * 128;
                    const int kr1 = ((kk + 1 < 153) ? kk + 1 : 152) * 128;
                    b.x = g2_Wso[kr0 + n];
                    b.y = g2_Wso[kr1 + n];
                }
                c8[nt] = __builtin_amdgcn_wmma_f32_16x16x4_f32(false, a, false, b,
                                                               (short)0, c8[nt], false, false);
            }
        }
#pragma unroll
        for (int nt = 0; nt < 8; ++nt) {
            const int n = nt * 16 + lc;
            const float bias = g2_bso[n];
#pragma unroll
            for (int r = 0; r < 8; ++r) {
                const int m = r + half * 8;
                float x = c8[nt][r] + bias;
                sm_s2[m][n] = x * sigm(x);                      // silu(s_out2)
            }
        }
    }
    __syncthreads();

    // ---- gate2 = sigmoid(silu(s_out2) @ Wg2 + bg2) via WMMA (4 split chains) ----
    {
        v8f part[4];
#pragma unroll
        for (int i = 0; i < 4; ++i) part[i] = vz;
        for (int kb = 0; kb < 128; kb += 16) {
#pragma unroll
            for (int i = 0; i < 4; ++i) {
                const int kk = kb + i * 4 + half * 2;
                v2f a = *(const v2f*)&sm_s2[lc][kk];
                v2f b;
                if (PACKED) {
                    b = *(const v2f*)&pw[PG2_OFF + ((kb >> 2) + i) * 64 + lc * 4 + half * 2];
                } else {
                    b.x = g2_Wg[kk * 16 + lc];
                    b.y = g2_Wg[(kk + 1) * 16 + lc];
                }
                part[i] = __builtin_amdgcn_wmma_f32_16x16x4_f32(false, a, false, b,
                                                                (short)0, part[i], false, false);
            }
        }
        v8f g8 = (part[0] + part[1]) + (part[2] + part[3]);
        const float bias = g2_bg[lc];
#pragma unroll
        for (int r = 0; r < 8; ++r) {
            const int m = r + half * 8;
            sm_gate[m][lc] = sigm(g8[r] + bias);
        }
    }
    __syncthreads();

    // ---- residual scalar: s = silu_s1 + silu_s2 (in merged2[:,0:128]) ----
    for (int t = lane; t < 16 * 128; t += 32) {
        int e = t >> 7, n = t & 127;
        sm_merged2[e][n] += sm_s2[e][n];
    }
    // ---- v_out2[48,16] = v_hid2 @ Wup2, gated + residual, atomic scatter v ----
    {
        v8f c3[3];
#pragma unroll
        for (int mt = 0; mt < 3; ++mt) c3[mt] = vz;
        for (int k0 = 0; k0 < 16; k0 += 4) {
            const int kk = k0 + half * 2;
            v2f b;
            b.x = g2_Wup[kk * 16 + lc];
            b.y = g2_Wup[(kk + 1) * 16 + lc];
#pragma unroll
            for (int mt = 0; mt < 3; ++mt) {
                v2f a = *(const v2f*)&vhid2_f[(mt * 16 + lc) * 16 + kk];
                c3[mt] = __builtin_amdgcn_wmma_f32_16x16x4_f32(false, a, false, b,
                                                               (short)0, c3[mt], false, false);
            }
        }
#pragma unroll
        for (int mt = 0; mt < 3; ++mt)
#pragma unroll
            for (int r = 0; r < 8; ++r) {
                int grow = mt * 16 + r + half * 8;
                int e = grow / 3, x = grow - e * 3;
                float vfin = c3[mt][r] * sm_gate[e][lc] + sm_vpre2[e][x][lc];
                atomicAdd(&out[(size_t)sm_row[e] * OUT_CH + 128 + lc * 3 + x], vfin);
            }
    }
    __syncthreads();

    // ---- scalar attention ----
    if (lane < 16) {
        float s = 0.0f;
        for (int n = 0; n < 128; ++n) s += sm_merged2[lane][n] * att_W[n];
        sm_attn[lane] = sigm(s + att_b[0]);
    }
    __syncthreads();
    // ---- scatter scalar part ----
    for (int t = lane; t < 16 * 128; t += 32) {
        int e = t >> 7, n = t & 127;
        atomicAdd(&out[(size_t)sm_row[e] * OUT_CH + n], sm_merged2[e][n] * sm_attn[e]);
    }
}

extern "C" void kernel_launch(void* const* d_in, const int* in_sizes, int n_in,
                              void* d_out, int out_size, void* d_ws, size_t ws_size,
                              hipStream_t stream) {
    (void)in_sizes; (void)n_in; (void)out_size;

    const float* node_s   = (const float*)d_in[0];
    const float* node_v   = (const float*)d_in[1];
    const float* edge_s   = (const float*)d_in[2];
    const float* edge_v   = (const float*)d_in[3];
    const float* frames   = (const float*)d_in[4];
    const float* g1_Wdown = (const float*)d_in[5];
    const float* g1_Wdf   = (const float*)d_in[6];
    const float* g1_Wso   = (const float*)d_in[7];
    const float* g1_bso   = (const float*)d_in[8];
    const float* g1_Wup   = (const float*)d_in[9];
    const float* g1_Wg    = (const float*)d_in[10];
    const float* g1_bg    = (const float*)d_in[11];
    const float* g2_Wdown = (const float*)d_in[12];
    const float* g2_Wdf   = (const float*)d_in[13];
    const float* g2_Wso   = (const float*)d_in[14];
    const float* g2_bso   = (const float*)d_in[15];
    const float* g2_Wup   = (const float*)d_in[16];
    const float* g2_Wg    = (const float*)d_in[17];
    const float* g2_bg    = (const float*)d_in[18];
    const float* att_W    = (const float*)d_in[19];
    const float* att_b    = (const float*)d_in[20];
    const int*   edge_idx = (const int*)d_in[21];
    float* out = (float*)d_out;
    float* pw  = (float*)d_ws;

    const bool packed = (pw != nullptr) && (ws_size >= (size_t)PW_BYTES);

    // segment_sum target must start from zero every call (harness poisons once)
    zero_kernel<<<(OUT_ELTS + 255) / 256, 256, 0, stream>>>(out, OUT_ELTS);

    const int n_tiles = N_EDGES / 16;   // 160000 % 16 == 0
    if (packed) {
        pack_weights<<<(PW_FLOATS + 255) / 256, 256, 0, stream>>>(
            g1_Wso, g2_Wso, g1_Wg, g2_Wg, pw);
        gcp_edge_kernel<true><<<n_tiles, 32, 0, stream>>>(
            node_s, node_v, edge_s, edge_v, frames,
            g1_Wdown, g1_Wdf, g1_Wso, g1_bso, g1_Wup, g1_Wg, g1_bg,
            g2_Wdown, g2_Wdf, g2_Wso, g2_bso, g2_Wup, g2_Wg, g2_bg,
            att_W, att_b, edge_idx, pw, out);
    } else {
        gcp_edge_kernel<false><<<n_tiles, 32, 0, stream>>>(
            node_s, node_v, edge_s, edge_v, frames,
            g1_Wdown, g1_Wdf, g1_Wso, g1_bso, g1_Wup, g1_Wg, g1_bg,
            g2_Wdown, g2_Wdf, g2_Wso, g2_bso, g2_Wup, g2_Wg, g2_bg,
            att_W, att_b, edge_idx, pw, out);
    }
}